// KnotForward_50113678410175
// MI455X (gfx1250) — compile-verified
//
#include <hip/hip_runtime.h>

// ---------------------------------------------------------------------------
// Piecewise-linear MLP knot propagation for MI455X (gfx1250, wave32, WMMA).
// One workgroup (8 wave32) per batch element; knot state + z matrix in LDS.
// Hidden-layer GEMMs (128x259 @ 259x256, f32) on v_wmma_f32_16x16x4_f32.
// h_in (relu(lerp(zL,zR,alpha)) + skip coords) is materialized in 32-wide
// K chunks into a small LDS staging buffer so the WMMA loop is branch-free:
// ds_load_b64 (A) + global_load_b32 x2 (B, L2-hot weights) + wmma.
// Each wave owns a 32-column N slice (disjoint W rows -> no redundant B
// traffic) and iterates all 8 M tiles.
// ---------------------------------------------------------------------------

typedef __attribute__((ext_vector_type(2))) float v2f;
typedef __attribute__((ext_vector_type(8))) float v8f;

#define KNOTS   128
#define NSEG    127        // KNOTS - 1
#define MCPS    8
#define HDIM    256
#define DDIM    3
#define KIN     259        // HDIM + DDIM
#define EPSF    1e-6f
#define ZPITCH  260        // 16B-aligned rows (260*4 % 16 == 0)
#define CHUNK   32         // K-chunk staged per materialization phase
#define HPITCH  36         // 16B-aligned rows, conflict-free A-frag banks

struct __align__(16) Shared {
  float zbuf[KNOTS][ZPITCH];   // current layer z (masked)      133,120 B
  float hst[KNOTS][HPITCH];    // staged h_in K-chunk             18,432 B
  float t_cur[KNOTS];
  int   seg_cur[KNOTS];        // descriptor: source segment row in zbuf
  float al_cur[KNOTS];         // descriptor: interpolation alpha
  float t_nxt[KNOTS];
  int   seg_nxt[KNOTS];
  float al_nxt[KNOTS];
  int   scan[KNOTS];
  float sp[DDIM];
  float dp[DDIM];
  int   red_ncross;
};

__device__ __forceinline__ float relu_lerp(float l, float r, float a) {
  float v = l + a * (r - l);
  return v > 0.0f ? v : 0.0f;
}

// ---------------------------------------------------------------------------
// GEMM: z[m][n] = b[n] + sum_k h_in[m][k] * W[n][k]   (M=128, N=256, K=259)
// Wave w owns columns n in [32w, 32w+32) (2 N-tiles), loops all 8 M-tiles.
// ---------------------------------------------------------------------------
__device__ __forceinline__ void gemm_layer(Shared& S,
                                           const float* __restrict__ W,
                                           const float* __restrict__ bvec) {
  const int tid  = threadIdx.x;
  const int wv   = tid >> 5;
  const int lane = tid & 31;
  const int hi   = lane >> 4;    // f32 A/B layout: lane half selects K pair
  const int l15  = lane & 15;

  // -- materialization context: this thread fills hst row mh, 16-col half ch0
  const int   mh   = tid >> 1;
  const int   ch0  = (tid & 1) << 4;
  const float tmh  = S.t_cur[mh];
  const bool  mvh  = (tmh <= 1.0f);
  const int   sh0  = S.seg_cur[mh];
  const int   sh1  = (sh0 + 1 < KNOTS) ? sh0 + 1 : sh0;
  const float ah   = S.al_cur[mh];
  const float* __restrict__ zlrow = S.zbuf[sh0];
  const float* __restrict__ zrrow = S.zbuf[sh1];
  float* __restrict__ hdst = &S.hst[mh][ch0];

  // -- wmma context: columns n0 (q=0) and n0+16 (q=1)
  const int n0 = (wv << 5) + l15;
  const float* __restrict__ wr0 = W + (size_t)n0 * KIN + hi * 2;
  const float* __restrict__ wr1 = W + (size_t)(n0 + 16) * KIN + hi * 2;

  // A-fragment row pointers (constant across chunks; hst is chunk-local)
  const float* hrow[8];
#pragma unroll
  for (int mt = 0; mt < 8; ++mt) hrow[mt] = &S.hst[mt * 16 + l15][hi * 2];

  v8f acc[8][2];
  {
    const float bv0 = bvec[n0];
    const float bv1 = bvec[n0 + 16];
#pragma unroll
    for (int mt = 0; mt < 8; ++mt)
#pragma unroll
      for (int j = 0; j < 8; ++j) { acc[mt][0][j] = bv0; acc[mt][1][j] = bv1; }
  }

  for (int c = 0; c < 8; ++c) {              // 8 chunks cover k = 0..255
    const int kc = c * CHUNK;
    __syncthreads();                         // hst free (prev chunk consumed)
    // ---- stage h_in[:, kc..kc+31] (branch-free interp+relu, float4 LDS) ----
#pragma unroll
    for (int cc = 0; cc < 16; cc += 4) {
      const int k = kc + ch0 + cc;
      float4 L = *(const float4*)(zlrow + k);
      float4 R = *(const float4*)(zrrow + k);
      float4 o;
      o.x = mvh ? relu_lerp(L.x, R.x, ah) : 0.0f;
      o.y = mvh ? relu_lerp(L.y, R.y, ah) : 0.0f;
      o.z = mvh ? relu_lerp(L.z, R.z, ah) : 0.0f;
      o.w = mvh ? relu_lerp(L.w, R.w, ah) : 0.0f;
      *(float4*)(hdst + cc) = o;
    }
    __syncthreads();
    // ---- 8 k-steps of 4: pure loads + wmma, no bounds checks ----
#pragma unroll 2
    for (int ks = 0; ks < 8; ++ks) {
      const int kg = kc + ks * 4;
      v2f bf0, bf1;
      bf0.x = wr0[kg]; bf0.y = wr0[kg + 1];
      bf1.x = wr1[kg]; bf1.y = wr1[kg + 1];
#pragma unroll
      for (int mt = 0; mt < 8; ++mt) {
        const v2f af = *(const v2f*)(hrow[mt] + ks * 4);
        acc[mt][0] = __builtin_amdgcn_wmma_f32_16x16x4_f32(
            false, af, false, bf0, (short)0, acc[mt][0], false, false);
        acc[mt][1] = __builtin_amdgcn_wmma_f32_16x16x4_f32(
            false, af, false, bf1, (short)0, acc[mt][1], false, false);
      }
    }
  }

  // ---- tail k-step (k = 256..259): skip coords from registers ----
  {
    v2f bf0, bf1;
    const int kg = 256;                       // wr* already offset by hi*2
    bf0.x = wr0[kg]; bf1.x = wr1[kg];         // k = 256 or 258, in bounds
    bf0.y = hi ? 0.0f : wr0[kg + 1];          // k = 257 valid, 259 -> 0
    bf1.y = hi ? 0.0f : wr1[kg + 1];
#pragma unroll
    for (int mt = 0; mt < 8; ++mt) {
      const int m = mt * 16 + l15;
      const float tm = S.t_cur[m];
      const bool  mv = (tm <= 1.0f);
      v2f af;
      af.x = mv ? (S.sp[hi * 2] + tm * S.dp[hi * 2]) : 0.0f;        // j=0|2
      af.y = (mv && !hi) ? (S.sp[1] + tm * S.dp[1]) : 0.0f;         // j=1|pad
      acc[mt][0] = __builtin_amdgcn_wmma_f32_16x16x4_f32(
          false, af, false, bf0, (short)0, acc[mt][0], false, false);
      acc[mt][1] = __builtin_amdgcn_wmma_f32_16x16x4_f32(
          false, af, false, bf1, (short)0, acc[mt][1], false, false);
    }
  }

  __syncthreads();   // all reads of zbuf (staging) complete before overwrite
#pragma unroll
  for (int mt = 0; mt < 8; ++mt) {
#pragma unroll
    for (int j = 0; j < 8; ++j) {
      const int m = mt * 16 + j + hi * 8;   // C layout: lanes>=16 -> M+8
      const bool mv = (S.t_cur[m] <= 1.0f);
      S.zbuf[m][n0]      = mv ? acc[mt][0][j] : 0.0f;
      S.zbuf[m][n0 + 16] = mv ? acc[mt][1][j] : 0.0f;
    }
  }
  __syncthreads();
}

// ---------------------------------------------------------------------------
// Zero-crossing insertion (matches lax.top_k first-8-by-channel + stable
// argsort by alpha + global 128-smallest-t truncation).
// ---------------------------------------------------------------------------
__device__ void insert_crossings(Shared& S, int* g_mins, int* g_mk, int call) {
  const int tid = threadIdx.x;

  if (tid < KNOTS) { S.t_nxt[tid] = 2.0f; S.seg_nxt[tid] = 0; S.al_nxt[tid] = 0.0f; }
  if (tid == 0) S.red_ncross = 0;
  __syncthreads();

  float alphas[MCPS];
  int m_ins = 0;
  int ncross = 0;
  if (tid < NSEG) {
    const float tl = S.t_cur[tid];
    const float tr = S.t_cur[tid + 1];
    if (tl <= 1.0f && tr <= 1.0f) {
      const float4* __restrict__ zl4 = (const float4*)S.zbuf[tid];
      const float4* __restrict__ zr4 = (const float4*)S.zbuf[tid + 1];
      for (int c4 = 0; c4 < HDIM / 4; ++c4) {
        const float4 L = zl4[c4];
        const float4 R = zr4[c4];
        float lv[4] = { L.x, L.y, L.z, L.w };
        float rv[4] = { R.x, R.y, R.z, R.w };
#pragma unroll
        for (int e = 0; e < 4; ++e) {
          const float den = rv[e] - lv[e];
          if (fabsf(den) > EPSF) {
            const float a = -lv[e] / den;
            if (a > EPSF && a < 1.0f - EPSF) {
              ++ncross;
              if (m_ins < MCPS) {            // first 8 by channel index
                int p = m_ins;
                while (p > 0 && alphas[p - 1] > a) { alphas[p] = alphas[p - 1]; --p; }
                alphas[p] = a;               // stable insertion by alpha
                ++m_ins;
              }
            }
          }
        }
      }
    }
  }
  if (ncross > 0) atomicMax(&S.red_ncross, ncross);

  int cnt = 0;
  if (tid < KNOTS) {
    cnt = (S.t_cur[tid] <= 1.0f) ? 1 : 0;
    cnt += m_ins;
    S.scan[tid] = cnt;
  }
  __syncthreads();

  for (int off = 1; off < KNOTS; off <<= 1) {   // inclusive Hillis-Steele
    int v = 0;
    if (tid < KNOTS) {
      v = S.scan[tid];
      if (tid >= off) v += S.scan[tid - off];
    }
    __syncthreads();
    if (tid < KNOTS) S.scan[tid] = v;
    __syncthreads();
  }

  if (tid < KNOTS) {
    const int pos = S.scan[tid] - cnt;
    const float tl = S.t_cur[tid];
    if (tl <= 1.0f && pos < KNOTS) {
      S.t_nxt[pos]   = tl;
      S.seg_nxt[pos] = tid;
      S.al_nxt[pos]  = 0.0f;
    }
    if (m_ins > 0) {
      const float dt = S.t_cur[tid + 1] - tl;
      for (int i = 0; i < m_ins; ++i) {
        const int p = pos + 1 + i;
        if (p < KNOTS) {
          S.t_nxt[p]   = tl + alphas[i] * dt;
          S.seg_nxt[p] = tid;
          S.al_nxt[p]  = alphas[i];
        }
      }
    }
  }
  __syncthreads();

  if (tid == 0) {
    atomicMax(&g_mins[call], S.red_ncross);
    const int total = S.scan[KNOTS - 1];
    atomicMax(&g_mk[call], total > KNOTS ? KNOTS : total);
  }
  if (tid < KNOTS) {
    S.t_cur[tid]   = S.t_nxt[tid];
    S.seg_cur[tid] = S.seg_nxt[tid];
    S.al_cur[tid]  = S.al_nxt[tid];
  }
  __syncthreads();
}

// ---------------------------------------------------------------------------
__global__ __launch_bounds__(256)
void knot_forward_kernel(const float* __restrict__ end_pts,
                         const float* __restrict__ start_pts,
                         const float* __restrict__ W0, const float* __restrict__ b0,
                         const float* __restrict__ W1, const float* __restrict__ b1,
                         const float* __restrict__ W2, const float* __restrict__ b2,
                         const float* __restrict__ W3, const float* __restrict__ b3,
                         const float* __restrict__ W4, const float* __restrict__ b4,
                         float* __restrict__ out_t,
                         float* __restrict__ out_valid,
                         float* __restrict__ out_y,
                         int* __restrict__ g_mins, int* __restrict__ g_mk) {
  __shared__ Shared S;
  const int b   = blockIdx.x;
  const int tid = threadIdx.x;

  if (tid < DDIM) {
    const float sv = start_pts[b * DDIM + tid];
    S.sp[tid] = sv;
    S.dp[tid] = end_pts[b * DDIM + tid] - sv;
  }
  if (tid < KNOTS) {
    S.t_cur[tid]   = (tid == 0) ? 0.0f : ((tid == 1) ? 1.0f : 2.0f);
    S.seg_cur[tid] = 0;
    S.al_cur[tid]  = 0.0f;
  }
  __syncthreads();

  // Layer 0: z[k][h] = alpha0[h]*t[k] + beta0[h] for k<2, else 0
  for (int idx = tid; idx < KNOTS * HDIM; idx += 256) {
    const int k = idx >> 8;
    const int h = idx & 255;
    float v = 0.0f;
    if (k < 2) {
      const float* __restrict__ wr = W0 + h * DDIM;
      float be = b0[h];
      float al = 0.0f;
#pragma unroll
      for (int j = 0; j < DDIM; ++j) { be += S.sp[j] * wr[j]; al += S.dp[j] * wr[j]; }
      v = (k == 0) ? be : (al + be);
    }
    S.zbuf[k][h] = v;
  }
  __syncthreads();

  insert_crossings(S, g_mins, g_mk, 0);

  const float* Ws[3] = { W1, W2, W3 };
  const float* bs[3] = { b1, b2, b3 };
  for (int li = 0; li < 3; ++li) {
    gemm_layer(S, Ws[li], bs[li]);
    insert_crossings(S, g_mins, g_mk, li + 1);
  }

  // Output layer: y = valid ? relu(h) @ W4.T + b4 : 0  (OUT=3, float4 VALU)
  if (tid < KNOTS) {
    const float t = S.t_cur[tid];
    const bool  v = (t <= 1.0f);
    float y0 = 0.0f, y1 = 0.0f, y2 = 0.0f;
    if (v) {
      const int   s  = S.seg_cur[tid];
      const int   s1 = (s + 1 < KNOTS) ? (s + 1) : s;
      const float a  = S.al_cur[tid];
      const float4* __restrict__ zl4 = (const float4*)S.zbuf[s];
      const float4* __restrict__ zr4 = (const float4*)S.zbuf[s1];
      const float4* __restrict__ w40 = (const float4*)(W4);
      const float4* __restrict__ w41 = (const float4*)(W4 + HDIM);
      const float4* __restrict__ w42 = (const float4*)(W4 + 2 * HDIM);
      for (int c4 = 0; c4 < HDIM / 4; ++c4) {
        const float4 L = zl4[c4];
        const float4 R = zr4[c4];
        float4 h4;
        h4.x = relu_lerp(L.x, R.x, a);
        h4.y = relu_lerp(L.y, R.y, a);
        h4.z = relu_lerp(L.z, R.z, a);
        h4.w = relu_lerp(L.w, R.w, a);
        const float4 wa = w40[c4], wb = w41[c4], wc = w42[c4];
        y0 += h4.x * wa.x + h4.y * wa.y + h4.z * wa.z + h4.w * wa.w;
        y1 += h4.x * wb.x + h4.y * wb.y + h4.z * wb.z + h4.w * wb.w;
        y2 += h4.x * wc.x + h4.y * wc.y + h4.z * wc.z + h4.w * wc.w;
      }
      y0 += b4[0]; y1 += b4[1]; y2 += b4[2];
    }
    out_t[b * KNOTS + tid]     = t;
    out_valid[b * KNOTS + tid] = v ? 1.0f : 0.0f;
    float* yp = out_y + ((size_t)b * KNOTS + tid) * 3;
    yp[0] = y0; yp[1] = y1; yp[2] = y2;
  }
}

__global__ void init_stats_kernel(int* __restrict__ ws) {
  if (threadIdx.x < 8) ws[threadIdx.x] = 0;
}

__global__ void finalize_stats_kernel(const int* __restrict__ ws,
                                      float* __restrict__ stats) {
  if (threadIdx.x == 0) {
    stats[0] = 2.0f;                       // mk[0] constant
    stats[1] = (float)ws[4];               // mk[1..4]
    stats[2] = (float)ws[5];
    stats[3] = (float)ws[6];
    stats[4] = (float)ws[7];
    stats[5] = (float)ws[0];               // mins[0..3]
    stats[6] = (float)ws[1];
    stats[7] = (float)ws[2];
    stats[8] = (float)ws[3];
  }
}

extern "C" void kernel_launch(void* const* d_in, const int* in_sizes, int n_in,
                              void* d_out, int out_size, void* d_ws, size_t ws_size,
                              hipStream_t stream) {
  const float* end_pts   = (const float*)d_in[0];
  const float* start_pts = (const float*)d_in[1];
  const float* W0 = (const float*)d_in[2];  const float* b0 = (const float*)d_in[3];
  const float* W1 = (const float*)d_in[4];  const float* b1 = (const float*)d_in[5];
  const float* W2 = (const float*)d_in[6];  const float* b2 = (const float*)d_in[7];
  const float* W3 = (const float*)d_in[8];  const float* b3 = (const float*)d_in[9];
  const float* W4 = (const float*)d_in[10]; const float* b4 = (const float*)d_in[11];

  const int B = in_sizes[0] / DDIM;   // 1024

  float* out       = (float*)d_out;
  float* out_t     = out;                                   // B*128
  float* out_valid = out + (size_t)B * KNOTS;               // B*128
  float* out_y     = out + (size_t)2 * B * KNOTS;           // B*128*3
  float* out_stats = out + (size_t)2 * B * KNOTS + (size_t)B * KNOTS * 3;

  int* ws = (int*)d_ws;   // [0..3]=mins max, [4..7]=mk max

  init_stats_kernel<<<1, 32, 0, stream>>>(ws);
  knot_forward_kernel<<<B, 256, 0, stream>>>(end_pts, start_pts,
                                             W0, b0, W1, b1, W2, b2, W3, b3, W4, b4,
                                             out_t, out_valid, out_y, ws, ws + 4);
  finalize_stats_kernel<<<1, 32, 0, stream>>>(ws, out_stats);
}